// KernelMix_14791867367799
// MI455X (gfx1250) — compile-verified
//
#include <hip/hip_runtime.h>

// ---------------------------------------------------------------------------
// Problem constants (from the reference): B=8, K=4, N=2048, D=3, IW=OW=32.
// d_out = [positions copy: 196608 f32][output_weights: 2097152 f32]
// d_ws  (floats): [invdeg2: 65536][pos4: 65536*4][scaled weights: 65536*32]
//                 total ~9.70 MB
// ---------------------------------------------------------------------------
#define N_        2048
#define IW_       32
#define NTOT      65536            /* 8*4*2048 nodes   */
#define POS_FLOATS 196608          /* NTOT * 3         */

typedef float v2f __attribute__((ext_vector_type(2)));
typedef float v8f __attribute__((ext_vector_type(8)));

// ---------------------------------------------------------------------------
// Kernel 1: per-node degree (full row sum of exp kernel), then write
//   invdeg2[n] = 1/deg^2,  pos4[n] = (x,y,z,|p|^2),  wscaled[n][w] = W*invdeg2,
// and copy positions into the first part of d_out.
// exp(-|pi-pj|^2/2) = exp(pi.pj - |pj|^2/2) * exp(-|pi|^2/2)  (factor i-term out)
// ---------------------------------------------------------------------------
__global__ __launch_bounds__(256) void prep_kernel(
    const float* __restrict__ pos, const float* __restrict__ wts,
    float* __restrict__ out_pos, float* __restrict__ invdeg2,
    float4* __restrict__ pos4, float* __restrict__ wscaled) {
  __shared__ float4 lds[256];
  const int tid  = threadIdx.x;
  const int node = blockIdx.x * 256 + tid;       // 256 consecutive nodes, same (b,k)
  const int nb   = (node >> 11) << 11;           // base node of this (b,k)

  const float px = pos[node * 3 + 0];
  const float py = pos[node * 3 + 1];
  const float pz = pos[node * 3 + 2];
  const float ri = px * px + py * py + pz * pz;

  float s = 0.0f;
  for (int j0 = 0; j0 < N_; j0 += 256) {
    __syncthreads();
    const int j = (nb + j0 + tid) * 3;
    const float qx = pos[j + 0], qy = pos[j + 1], qz = pos[j + 2];
    lds[tid] = make_float4(qx, qy, qz, qx * qx + qy * qy + qz * qz);
    __syncthreads();
#pragma unroll 8
    for (int t = 0; t < 256; ++t) {
      const float4 q = lds[t];
      s += __expf(px * q.x + py * q.y + pz * q.z - 0.5f * q.w);
    }
  }
  const float deg = __expf(-0.5f * ri) * s;
  const float iv  = 1.0f / (deg * deg);

  invdeg2[node] = iv;
  pos4[node]    = make_float4(px, py, pz, ri);
  out_pos[node * 3 + 0] = px;
  out_pos[node * 3 + 1] = py;
  out_pos[node * 3 + 2] = pz;

  const float4* w4  = (const float4*)wts + node * 8;
  float4*       ws4 = (float4*)wscaled + node * 8;
#pragma unroll
  for (int t = 0; t < 8; ++t) {
    const float4 v = w4[t];
    ws4[t] = make_float4(v.x * iv, v.y * iv, v.z * iv, v.w * iv);
  }
}

// ---------------------------------------------------------------------------
// Kernel 2: one wave32 per 16-row strip of one (b,k).
//   acc[16x32] = sum_j exp(pi.pj - (ri+rj)/2) * wscaled[j][:]   via
//   V_WMMA_F32_16X16X4_F32 (K-tile built directly in A-register layout),
//   then row-scale by invdeg2[i], then 32x32 linear layer via WMMA (LDS
//   relayout C->A), bias add, store.
// ---------------------------------------------------------------------------
__global__ __launch_bounds__(32) void rbf_wmma_kernel(
    const float* __restrict__ invdeg2, const float4* __restrict__ pos4,
    const float* __restrict__ wscaled, const float* __restrict__ lin_w,
    const float* __restrict__ lin_b, float* __restrict__ out) {
  __shared__ float ldsO[16 * IW_];

  const int l  = threadIdx.x;
  const int m  = l & 15;          // A: row, B: col, C: col
  const int hi = l >> 4;          // lane half
  const int k0 = hi << 1;         // A/B K-slot: lanes 0-15 -> {0,1}, 16-31 -> {2,3}
  const int bk = blockIdx.x >> 7;
  const int i0 = (blockIdx.x & 127) << 4;
  const int nb = bk << 11;

  const float4 pI = pos4[nb + i0 + m];  // strip row m position + |p|^2

  v8f acc0 = {};  // cols 0..15 of W
  v8f acc1 = {};  // cols 16..31 of W

  for (int jt = 0; jt < 128; ++jt) {
    const int jb = nb + (jt << 4);
#pragma unroll
    for (int c = 0; c < 4; ++c) {
      const int jA = jb + (c << 2) + k0;        // this lane's two K-columns
      const float4 q0 = pos4[jA];
      const float4 q1 = pos4[jA + 1];
      v2f a;
      a.x = __expf(pI.x * q0.x + pI.y * q0.y + pI.z * q0.z - 0.5f * (pI.w + q0.w));
      a.y = __expf(pI.x * q1.x + pI.y * q1.y + pI.z * q1.z - 0.5f * (pI.w + q1.w));

      const float* r0 = wscaled + ((size_t)jA << 5);
      const float* r1 = wscaled + ((size_t)(jA + 1) << 5);
      v2f b0, b1;
      b0.x = r0[m];      b0.y = r1[m];
      b1.x = r0[m + 16]; b1.y = r1[m + 16];

      acc0 = __builtin_amdgcn_wmma_f32_16x16x4_f32(false, a, false, b0,
                                                   (short)0, acc0, false, false);
      acc1 = __builtin_amdgcn_wmma_f32_16x16x4_f32(false, a, false, b1,
                                                   (short)0, acc1, false, false);
    }
  }

  // Row-scale by invdeg2[i]; spill scaled 16x32 tile to LDS row-major.
  // C layout: vgpr r, lane l -> row hi*8+r, col m (acc0) / 16+m (acc1).
#pragma unroll
  for (int r = 0; r < 8; ++r) {
    const int row = (hi << 3) + r;
    const float iv = invdeg2[nb + i0 + row];
    ldsO[row * IW_ + m]      = acc0[r] * iv;
    ldsO[row * IW_ + 16 + m] = acc1[r] * iv;
  }
  __syncthreads();

  // Linear layer: D[i][o] = sum_w O[i][w] * lin_w[o][w]  (B = lin_w^T)
  v8f d0 = {};
  v8f d1 = {};
#pragma unroll
  for (int c = 0; c < 8; ++c) {
    const int kk = (c << 2) + k0;
    v2f a, b0, b1;
    a.x  = ldsO[m * IW_ + kk];          a.y  = ldsO[m * IW_ + kk + 1];
    b0.x = lin_w[m * IW_ + kk];         b0.y = lin_w[m * IW_ + kk + 1];
    b1.x = lin_w[(16 + m) * IW_ + kk];  b1.y = lin_w[(16 + m) * IW_ + kk + 1];
    d0 = __builtin_amdgcn_wmma_f32_16x16x4_f32(false, a, false, b0,
                                               (short)0, d0, false, false);
    d1 = __builtin_amdgcn_wmma_f32_16x16x4_f32(false, a, false, b1,
                                               (short)0, d1, false, false);
  }

  const float lb0 = lin_b[m];
  const float lb1 = lin_b[16 + m];
  float* op = out + POS_FLOATS;
#pragma unroll
  for (int r = 0; r < 8; ++r) {
    const int i = i0 + (hi << 3) + r;
    op[((size_t)(nb + i) << 5) + m]      = d0[r] + lb0;
    op[((size_t)(nb + i) << 5) + 16 + m] = d1[r] + lb1;
  }
}

// ---------------------------------------------------------------------------
extern "C" void kernel_launch(void* const* d_in, const int* in_sizes, int n_in,
                              void* d_out, int out_size, void* d_ws, size_t ws_size,
                              hipStream_t stream) {
  const float* positions = (const float*)d_in[0];
  const float* weights   = (const float*)d_in[1];
  const float* lin_w     = (const float*)d_in[2];
  const float* lin_b     = (const float*)d_in[3];
  float* out = (float*)d_out;

  float*  wsf     = (float*)d_ws;
  float*  invdeg2 = wsf;                    // NTOT floats
  float4* pos4    = (float4*)(wsf + NTOT);  // NTOT float4
  float*  wscaled = wsf + 5 * NTOT;         // NTOT * 32 floats

  prep_kernel<<<NTOT / 256, 256, 0, stream>>>(positions, weights, out,
                                              invdeg2, pos4, wscaled);
  rbf_wmma_kernel<<<NTOT / 16, 32, 0, stream>>>(invdeg2, pos4, wscaled,
                                                lin_w, lin_b, out);
}